// TFTest_54099408060639
// MI455X (gfx1250) — compile-verified
//
#include <hip/hip_runtime.h>
#include <math.h>

#define H_  256
#define N_  32
#define L_  8192
#define M_  4096   // L/2, complex ifft size = 16^3

typedef __attribute__((ext_vector_type(2))) float v2f;
typedef __attribute__((ext_vector_type(8))) float v8f;

#define TWO_PI 6.28318530717958647692f

// fast hardware reciprocal (v_rcp_f32, ~1 ulp)
__device__ __forceinline__ float frcp(float x) { return __builtin_amdgcn_rcpf(x); }

// LDS partition offsets (bytes)
//  sKf  : float2[4104]  (4097 used; also FFT "pong" buffer)   @ 0
//  sBuf : float2[4096]  (Z / FFT "ping" buffer)               @ 32832
//  sVr  : float[4*32]                                         @ 65600
//  sVi  : float[4*32]                                         @ 66112
//  sWdt : float2[32]                                          @ 66624
//  sW16 : float2[16]                                          @ 66880
#define SMEM_BYTES 67008

__global__ __launch_bounds__(256)
void s4_nplr_kernel(const float* __restrict__ w_re, const float* __restrict__ w_im,
                    const float* __restrict__ p_re, const float* __restrict__ p_im,
                    const float* __restrict__ B_re, const float* __restrict__ B_im,
                    const float* __restrict__ C_re, const float* __restrict__ C_im,
                    const float* __restrict__ log_dt, float* __restrict__ out)
{
    extern __shared__ unsigned char smem[];
    float2* sKf  = (float2*)(smem);
    float2* sBuf = (float2*)(smem + 32832);
    float*  sVr  = (float*) (smem + 65600);
    float*  sVi  = (float*) (smem + 66112);
    float2* sWdt = (float2*)(smem + 66624);
    float2* sW16 = (float2*)(smem + 66880);

    const int h    = blockIdx.x;
    const int tid  = threadIdx.x;
    const int lane = tid & 31;
    const int wave = tid >> 5;
    const float dt = __expf(log_dt[h]);

    // ---- init tables: v combos, dt-scaled poles, W16 twiddle table ----
    if (tid < N_) {
        const int n = tid;
        const int idx = h * N_ + n;
        const float br = B_re[idx], bi = B_im[idx];
        const float pr = p_re[idx], pi = p_im[idx];
        const float cr = C_re[idx], ci = C_im[idx];
        sVr[0*32+n] = br*cr - bi*ci;  sVi[0*32+n] = br*ci + bi*cr;   // B*C
        sVr[1*32+n] = br*pr + bi*pi;  sVi[1*32+n] = bi*pr - br*pi;   // B*conj(P)
        sVr[2*32+n] = pr*cr - pi*ci;  sVi[2*32+n] = pr*ci + pi*cr;   // P*C
        sVr[3*32+n] = pr*pr + pi*pi;  sVi[3*32+n] = 0.0f;            // P*conj(P)
        sWdt[n] = make_float2(w_re[idx]*dt, w_im[idx]*dt);
    }
    if (tid < 16) {
        float s, c;
        __sincosf(TWO_PI * (float)tid * (1.0f/16.0f), &s, &c);
        sW16[tid] = make_float2(c, s);   // e^{+2*pi*i*k/16}
    }
    __syncthreads();

    // ---- Stage A: Cauchy sum + Woodbury, per-h GEMM via V_WMMA_F32_16X16X4_F32 ----
    // A (16x4, rows 0..7 used): per pole n, rows = Re/Im of r00,r01,r10,r11,
    //   cols k = coefficient of (cw_re, cw_im, cwc_re, cwc_im):
    //   Re row: [vr, -vi, vr,  vi];  Im row: [vi, vr, -vi, vr]
    // B (4x16): row k per 16 frequencies. Lanes 0-15 compute cw = 1/(z-w),
    //   lanes 16-31 compute cwc = 1/(z-conj(w)) for the same 16 freqs.
    const int   m      = lane & 15;
    const bool  hihalf = lane >= 16;
    const bool  act    = m < 8;
    const bool  reRow  = (m & 1) == 0;
    const int   sc     = (m >> 1) & 3;
    const float sA0    = hihalf ? -1.0f : 1.0f;  // sign of vi in a0 for Im rows
    const float sA1    = hihalf ?  1.0f : -1.0f; // sign of vi in a1 for Re rows
    const float wSgn   = hihalf ? -1.0f : 1.0f;  // conj(w) for hi lanes

    // Hoist tile-invariant per-lane data into registers:
    //   A fragments (a0,a1 per pole) and effective pole (wnx, wny with conj sign).
    float a0v[N_], a1v[N_], wnx[N_], wny[N_];
#pragma unroll
    for (int n = 0; n < N_; ++n) {
        float vr = sVr[sc*32 + n];          // unconditional LDS load (address valid for all lanes)
        float vi = sVi[sc*32 + n];
        vr = act ? vr : 0.0f;
        vi = act ? vi : 0.0f;
        a0v[n] = reRow ? vr       : sA0 * vi;
        a1v[n] = reRow ? sA1 * vi : vr;
        const float2 wn = sWdt[n];
        wnx[n] = wn.x;
        wny[n] = wSgn * wn.y;
    }

    for (int T = wave; T < 256; T += 8) {
        const int l = T * 16 + m;
        float so, co;
        __sincosf(-TWO_PI * (float)l * (1.0f/8192.0f), &so, &co);
        // omega = (co, so); den = 1+omega; z = 2(1-omega)/(1+omega)
        const float denr = 1.0f + co, deni = so;
        const float dinv = frcp(denr*denr + deni*deni);
        const float numr = 2.0f * (1.0f - co), numi = -2.0f * so;
        const float zr = (numr*denr + numi*deni) * dinv;
        const float zi = (numi*denr - numr*deni) * dinv;

        v8f acc = {0.f,0.f,0.f,0.f,0.f,0.f,0.f,0.f};
#pragma unroll
        for (int n = 0; n < N_; ++n) {
            const float dre  = zr - wnx[n];
            const float dim  = zi - wny[n];
            const float rinv = frcp(fmaf(dre, dre, dim*dim));
            v2f b;
            b.x =  dre * rinv;      // Re 1/(z-w)   (rows K0 / K2)
            b.y = -dim * rinv;      // Im 1/(z-w)   (rows K1 / K3)
            v2f a;
            a.x = a0v[n];
            a.y = a1v[n];
            acc = __builtin_amdgcn_wmma_f32_16x16x4_f32(
                      false, a, false, b, (short)0, acc, false, false);
        }
        if (!hihalf) {
            // lanes 0-15: acc[g] = r-component g for frequency l
            const float r00r = acc[0]*dt, r00i = acc[1]*dt;
            const float r01r = acc[2]*dt, r01i = acc[3]*dt;
            const float r10r = acc[4]*dt, r10i = acc[5]*dt;
            const float r11r = acc[6]*dt, r11i = acc[7]*dt;
            const float ddr = 1.0f + r11r, ddi = r11i;
            const float dd  = frcp(fmaf(ddr, ddr, ddi*ddi));
            const float nr  = r01r*r10r - r01i*r10i;
            const float ni  = r01r*r10i + r01i*r10r;
            const float qr  = (nr*ddr + ni*ddi) * dd;
            const float qi  = (ni*ddr - nr*ddi) * dd;
            const float kr  = r00r - qr, ki = r00i - qi;
            const float fr  =  2.0f * denr * dinv;   // 2/(1+omega)
            const float fi  = -2.0f * deni * dinv;
            sKf[l] = make_float2(kr*fr - ki*fi, kr*fi + ki*fr);
        }
    }

    // ---- Nyquist bin l = 4096 (scalar, one thread) ----
    if (tid == 0) {
        float so, co;
        __sincosf(-TWO_PI * (float)M_ * (1.0f/8192.0f), &so, &co);
        const float denr = 1.0f + co, deni = so;
        const float dinv = frcp(denr*denr + deni*deni);
        const float numr = 2.0f * (1.0f - co), numi = -2.0f * so;
        const float zr = (numr*denr + numi*deni) * dinv;
        const float zi = (numi*denr - numr*deni) * dinv;
        float rr[4] = {0,0,0,0}, ri[4] = {0,0,0,0};
        for (int n = 0; n < N_; ++n) {
            const float2 wn = sWdt[n];
            float dre = zr - wn.x, dim = zi - wn.y;
            float rin = frcp(fmaf(dre, dre, dim*dim));
            const float cwr = dre*rin, cwi = -dim*rin;
            dim = zi + wn.y;
            rin = frcp(fmaf(dre, dre, dim*dim));
            const float ccr = dre*rin, cci = -dim*rin;
            for (int s = 0; s < 4; ++s) {
                const float vr = sVr[s*32+n], vi = sVi[s*32+n];
                rr[s] += vr*cwr - vi*cwi + vr*ccr + vi*cci;
                ri[s] += vr*cwi + vi*cwr + vr*cci - vi*ccr;
            }
        }
        const float r00r = rr[0]*dt, r00i = ri[0]*dt;
        const float r01r = rr[1]*dt, r01i = ri[1]*dt;
        const float r10r = rr[2]*dt, r10i = ri[2]*dt;
        const float r11r = rr[3]*dt, r11i = ri[3]*dt;
        const float ddr = 1.0f + r11r, ddi = r11i;
        const float dd  = frcp(fmaf(ddr, ddr, ddi*ddi));
        const float nr  = r01r*r10r - r01i*r10i;
        const float ni  = r01r*r10i + r01i*r10r;
        const float qr  = (nr*ddr + ni*ddi) * dd;
        const float qi  = (ni*ddr - nr*ddi) * dd;
        const float kr  = r00r - qr, ki = r00i - qi;
        const float fr  =  2.0f * denr * dinv;
        const float fi  = -2.0f * deni * dinv;
        sKf[M_] = make_float2(kr*fr - ki*fi, kr*fi + ki*fr);
    }
    __syncthreads();

    // ---- Stage B: irfft(8192) via half-size complex ifft(4096 = 16^3) ----
    // Pack: Z[k] = E[k] + i*O[k];  E=(X[k]+conj(X[M-k]))/2, O=(X[k]-conj(X[M-k]))*e^{+2pi i k/L}/2
    for (int k = tid; k < M_; k += 256) {
        const float2 X0 = sKf[k];
        const float2 X1 = sKf[M_ - k];
        const float x1r = X1.x, x1i = -X1.y;
        const float Er = 0.5f*(X0.x + x1r), Ei = 0.5f*(X0.y + x1i);
        const float Dr = 0.5f*(X0.x - x1r), Di = 0.5f*(X0.y - x1i);
        float s, c;
        __sincosf(TWO_PI * (float)k * (1.0f/8192.0f), &s, &c);
        const float Or = Dr*c - Di*s, Oi = Dr*s + Di*c;
        sBuf[k] = make_float2(Er - Oi, Ei + Or);   // E + i*O
    }
    __syncthreads();

    float xr[16], xi[16];

    // Stage 1: sum over m2 (stride 256), no twiddle.  sBuf -> sKf
    {
        const int base = tid;
#pragma unroll
        for (int j = 0; j < 16; ++j) { float2 v = sBuf[base + 256*j]; xr[j] = v.x; xi[j] = v.y; }
        for (int t = 0; t < 16; ++t) {
            float ar = 0.f, ai = 0.f;
#pragma unroll
            for (int j = 0; j < 16; ++j) {
                const float2 w = sW16[(j*t) & 15];
                ar += xr[j]*w.x - xi[j]*w.y;
                ai += xr[j]*w.y + xi[j]*w.x;
            }
            sKf[base + 256*t] = make_float2(ar, ai);
        }
    }
    __syncthreads();

    // Stage 2: twiddle e^{+2pi i m1 t2/256}, sum over m1 (stride 16).  sKf -> sBuf
    {
        const int m0 = tid & 15, t2 = tid >> 4;
#pragma unroll
        for (int j = 0; j < 16; ++j) {
            const float2 v = sKf[m0 + 16*j + 256*t2];
            const int idx = (j * t2) & 255;
            float s, c;
            __sincosf(TWO_PI * (float)idx * (1.0f/256.0f), &s, &c);
            xr[j] = v.x*c - v.y*s;
            xi[j] = v.x*s + v.y*c;
        }
        for (int t = 0; t < 16; ++t) {
            float ar = 0.f, ai = 0.f;
#pragma unroll
            for (int j = 0; j < 16; ++j) {
                const float2 w = sW16[(j*t) & 15];
                ar += xr[j]*w.x - xi[j]*w.y;
                ai += xr[j]*w.y + xi[j]*w.x;
            }
            sBuf[m0 + 16*t + 256*t2] = make_float2(ar, ai);
        }
    }
    __syncthreads();

    // Stage 3: twiddle e^{+2pi i m0 (t2+16 t1)/4096}, sum over m0 (stride 1).  sBuf -> sKf
    {
        const int t1 = tid & 15, t2 = tid >> 4;
        const int tt = t2 + 16*t1;
#pragma unroll
        for (int j = 0; j < 16; ++j) {
            const float2 v = sBuf[j + 16*t1 + 256*t2];
            const int idx = (j * tt) & 4095;
            float s, c;
            __sincosf(TWO_PI * (float)idx * (1.0f/4096.0f), &s, &c);
            xr[j] = v.x*c - v.y*s;
            xi[j] = v.x*s + v.y*c;
        }
        for (int t = 0; t < 16; ++t) {
            float ar = 0.f, ai = 0.f;
#pragma unroll
            for (int j = 0; j < 16; ++j) {
                const float2 w = sW16[(j*t) & 15];
                ar += xr[j]*w.x - xi[j]*w.y;
                ai += xr[j]*w.y + xi[j]*w.x;
            }
            sKf[t2 + 16*t1 + 256*t] = make_float2(ar, ai);
        }
    }
    __syncthreads();

    // Unpack: x[2m] = Re y[m] / 4096, x[2m+1] = Im y[m] / 4096
    {
        const float scale = 1.0f / 4096.0f;
        float2* op = (float2*)(out + (size_t)h * L_);
        for (int mm = tid; mm < M_; mm += 256) {
            const float2 y = sKf[mm];
            op[mm] = make_float2(y.x * scale, y.y * scale);
        }
    }
}

extern "C" void kernel_launch(void* const* d_in, const int* in_sizes, int n_in,
                              void* d_out, int out_size, void* d_ws, size_t ws_size,
                              hipStream_t stream) {
    (void)in_sizes; (void)n_in; (void)out_size; (void)d_ws; (void)ws_size;
    const float* w_re   = (const float*)d_in[0];
    const float* w_im   = (const float*)d_in[1];
    const float* p_re   = (const float*)d_in[2];
    const float* p_im   = (const float*)d_in[3];
    const float* B_re   = (const float*)d_in[4];
    const float* B_im   = (const float*)d_in[5];
    const float* C_re   = (const float*)d_in[6];
    const float* C_im   = (const float*)d_in[7];
    const float* log_dt = (const float*)d_in[8];
    float* out = (float*)d_out;

    s4_nplr_kernel<<<H_, 256, SMEM_BYTES, stream>>>(
        w_re, w_im, p_re, p_im, B_re, B_im, C_re, C_im, log_dt, out);
}